// RankSoftmaxContrastiveLearning_33440615367032
// MI455X (gfx1250) — compile-verified
//
#include <hip/hip_runtime.h>

// ---------------------------------------------------------------------------
// Decoupled contrastive loss on MI455X (gfx1250, wave32).
//   sims = (A @ T^T) * exp(temp);  E = exp(sims)
//   num  = diag(E); den_i = rowsum(E, diag=0); den_j = colsum(E, diag=0)
//   loss = mean(-log num + 0.5*(log den_i + log den_j))
// GEMM via V_WMMA_F32_16X16X4_F32. Panels staged in LDS in PAIRED fragment
// order: two adjacent 16x4 fragments interleaved per lane (dword addr =
// pair*132 + 4*lane), so one ds_load_b128 yields TWO WMMA fragments and all
// offsets fold into the 16-bit ds immediate. Stride 132 = 4 (mod 64) keeps
// staging stores bank-conflict-free. Two independent WMMA accumulator chains.
// ---------------------------------------------------------------------------

typedef float v2f __attribute__((ext_vector_type(2)));
typedef float v4f __attribute__((ext_vector_type(4)));
typedef float v8f __attribute__((ext_vector_type(8)));

#define D_DIM 128          // latent dim (fixed by reference)
#define BT    128          // block tile: 128 rows x 128 cols of sims
#define PS    132          // dwords per fragment PAIR (2*64 data + 4 pad)
#define PANEL ((BT / 16) * 16 * PS)   // 16896 floats = 67584 B per panel

#if __has_builtin(__builtin_amdgcn_exp2f)
#define EXP2F(x) __builtin_amdgcn_exp2f(x)
#else
#define EXP2F(x) exp2f(x)
#endif

#define LO2(x) __builtin_shufflevector((x), (x), 0, 1)
#define HI2(x) __builtin_shufflevector((x), (x), 2, 3)

// ---------------------------------------------------------------------------
__global__ void dcl_zero_kernel(float* __restrict__ p, int n) {
    int i = blockIdx.x * blockDim.x + threadIdx.x;
    if (i < n) p[i] = 0.0f;
}

// ---------------------------------------------------------------------------
// Per-subtile epilogue: exp, diagonal extract/mask, row/col partial sums.
__device__ __forceinline__ void dcl_epilogue(
    v8f acc, int ct, bool onDiagBlock, int wave, int lane, int r0, float s2,
    float* __restrict__ num, float (&rowpart)[8], float* s_col)
{
    const int n = lane & 15;            // C/D layout: N = lane%16
    const int g = lane >> 4;            // half: M = v + 8*g
    const bool diagTile = onDiagBlock && (ct == wave);
    float csum = 0.0f;
#pragma unroll
    for (int v = 0; v < 8; ++v) {
        float val = EXP2F(acc[v] * s2);       // exp(scale*sim) via v_exp_f32
        if (diagTile && (v + 8 * g) == n) {   // global row == global col
            num[r0 + wave * 16 + n] = val;    // numerator (unmasked)
            val = 0.0f;                       // decoupled: drop from denominators
        }
        rowpart[v] += val;
        csum += val;                          // col partial over m = g*8..g*8+7
    }
    csum += __shfl_xor(csum, 16, 32);         // combine halves -> 16-row col sum
    if (lane < 16) atomicAdd(&s_col[ct * 16 + n], csum);
}

// ---------------------------------------------------------------------------
// One block = 8 waves computes a BTxBT tile of exp_sims.
// Paired fragment layout (per 128x128 fp32 panel):
//   pair pi = rt*16 + pr   (rt: 16-row group 0..7; pr = kg/2, kg: K-group of 4)
//   lane L = m + 16*g holds dwords [pi*PS + 4L .. +3] =
//     { A[m][8pr+2g], A[m][8pr+2g+1], A[m][8pr+4+2g], A[m][8pr+4+2g+1] }
//   i.e. v0/v1 of fragment kg=2pr, then v0/v1 of fragment kg=2pr+1.
__global__ __launch_bounds__(256) void dcl_tile_kernel(
    const float* __restrict__ A, const float* __restrict__ T,
    const float* __restrict__ temp,
    float* __restrict__ num, float* __restrict__ den_i, float* __restrict__ den_j)
{
    __shared__ float lA[PANEL];
    __shared__ float lB[PANEL];
    __shared__ float s_row[BT];
    __shared__ float s_col[BT];

    const int tid  = threadIdx.x;
    const int lane = tid & 31;
    const int wave = tid >> 5;
    const int r0   = blockIdx.y * BT;   // audio (row) block
    const int c0   = blockIdx.x * BT;   // text  (col) block

    if (tid < BT) s_row[tid] = 0.0f;
    else          s_col[tid - BT] = 0.0f;

    // ---- stage both panels: thread handles 8-float row chunks.
    // pass p: chunk = tid + 256p; row = tid/16 + 16p (rt = p, m = tid/16),
    // pr = tid%16. Store addr = (p*16 + pr)*PS + 4m  -> base + immediate.
    {
        const int pr = tid & 15;
        const int m  = tid >> 4;
        const float4* gA = reinterpret_cast<const float4*>(A + (size_t)r0 * D_DIM);
        const float4* gT = reinterpret_cast<const float4*>(T + (size_t)c0 * D_DIM);
        float* sA = &lA[pr * PS + 4 * m];
        float* sB = &lB[pr * PS + 4 * m];
#pragma unroll
        for (int p = 0; p < 8; ++p) {
            const int soff = p * 16 * PS;                 // dwords
            float4 va0 = gA[2 * tid + 512 * p];           // k = 8pr .. 8pr+3
            float4 va1 = gA[2 * tid + 512 * p + 1];       // k = 8pr+4 .. 8pr+7
            float4 vt0 = gT[2 * tid + 512 * p];
            float4 vt1 = gT[2 * tid + 512 * p + 1];
            // lane m   (g=0): {k0, k1, k4, k5};  lane m+16 (g=1): {k2, k3, k6, k7}
            *reinterpret_cast<v4f*>(sA + soff)      = (v4f){va0.x, va0.y, va1.x, va1.y};
            *reinterpret_cast<v4f*>(sA + soff + 64) = (v4f){va0.z, va0.w, va1.z, va1.w};
            *reinterpret_cast<v4f*>(sB + soff)      = (v4f){vt0.x, vt0.y, vt1.x, vt1.y};
            *reinterpret_cast<v4f*>(sB + soff + 64) = (v4f){vt0.z, vt0.w, vt1.z, vt1.w};
        }
    }
    __syncthreads();

    // ---- wave w owns rows [w*16, w*16+16): load its 16 fragment pairs once
    v4f afr[16];
    {
        const float* ap = &lA[wave * 16 * PS + 4 * lane];
#pragma unroll
        for (int pr = 0; pr < 16; ++pr)
            afr[pr] = *reinterpret_cast<const v4f*>(ap + pr * PS);
    }

    // exp(scale*sim) = exp2(sim * scale*log2e); scale = exp(temp[0])
    const float s2 = __expf(temp[0]) * 1.44269504088896f;
    const bool onDiagBlock = (r0 == c0);
    float rowpart[8] = {0, 0, 0, 0, 0, 0, 0, 0};

    // ---- two independent accumulator chains (col sub-tiles ct0, ct1)
#pragma unroll
    for (int ctp = 0; ctp < (BT / 16) / 2; ++ctp) {
        const float* bq = &lB[ctp * 32 * PS + 4 * lane];  // base for this ct pair
        v8f acc0 = {0.f, 0.f, 0.f, 0.f, 0.f, 0.f, 0.f, 0.f};
        v8f acc1 = {0.f, 0.f, 0.f, 0.f, 0.f, 0.f, 0.f, 0.f};
#pragma unroll
        for (int pr = 0; pr < 16; ++pr) {
            v4f b0 = *reinterpret_cast<const v4f*>(bq + pr * PS);         // ct0
            v4f b1 = *reinterpret_cast<const v4f*>(bq + (16 + pr) * PS);  // ct1
            v2f a0 = LO2(afr[pr]), a1 = HI2(afr[pr]);
            // (neg_a, A, neg_b, B, c_mod, C, reuse_a, reuse_b)
            acc0 = __builtin_amdgcn_wmma_f32_16x16x4_f32(
                false, a0, false, LO2(b0), (short)0, acc0, false, false);
            acc1 = __builtin_amdgcn_wmma_f32_16x16x4_f32(
                false, a0, false, LO2(b1), (short)0, acc1, false, false);
            acc0 = __builtin_amdgcn_wmma_f32_16x16x4_f32(
                false, a1, false, HI2(b0), (short)0, acc0, false, false);
            acc1 = __builtin_amdgcn_wmma_f32_16x16x4_f32(
                false, a1, false, HI2(b1), (short)0, acc1, false, false);
        }
        dcl_epilogue(acc0, 2 * ctp,     onDiagBlock, wave, lane, r0, s2, num, rowpart, s_col);
        dcl_epilogue(acc1, 2 * ctp + 1, onDiagBlock, wave, lane, r0, s2, num, rowpart, s_col);
    }

    // ---- row sums: reduce over the 16 N-lanes in each half
#pragma unroll
    for (int v = 0; v < 8; ++v) {
        float r = rowpart[v];
        r += __shfl_xor(r, 1, 32);
        r += __shfl_xor(r, 2, 32);
        r += __shfl_xor(r, 4, 32);
        r += __shfl_xor(r, 8, 32);
        if ((lane & 15) == 0)
            atomicAdd(&s_row[wave * 16 + v + 8 * (lane >> 4)], r);
    }
    __syncthreads();

    if (tid < BT) atomicAdd(&den_i[r0 + tid], s_row[tid]);
    else          atomicAdd(&den_j[c0 + (tid - BT)], s_col[tid - BT]);
}

// ---------------------------------------------------------------------------
__global__ __launch_bounds__(256) void dcl_finalize_kernel(
    const float* __restrict__ num, const float* __restrict__ den_i,
    const float* __restrict__ den_j, float* __restrict__ out, int n)
{
    const int tid = threadIdx.x;
    float acc = 0.0f;
    for (int i = tid; i < n; i += 256) {
        float nu = fmaxf(num[i],   1e-20f);
        float di = fmaxf(den_i[i], 1e-20f);
        float dj = fmaxf(den_j[i], 1e-20f);
        acc += -logf(nu) + 0.5f * (logf(di) + logf(dj));
    }
    acc += __shfl_xor(acc, 1, 32);
    acc += __shfl_xor(acc, 2, 32);
    acc += __shfl_xor(acc, 4, 32);
    acc += __shfl_xor(acc, 8, 32);
    acc += __shfl_xor(acc, 16, 32);
    __shared__ float red[8];
    if ((tid & 31) == 0) red[tid >> 5] = acc;
    __syncthreads();
    if (tid == 0) {
        float t = 0.0f;
        for (int w = 0; w < 8; ++w) t += red[w];
        out[0] = t / (float)n;
    }
}

// ---------------------------------------------------------------------------
extern "C" void kernel_launch(void* const* d_in, const int* in_sizes, int n_in,
                              void* d_out, int out_size, void* d_ws, size_t ws_size,
                              hipStream_t stream) {
    (void)n_in; (void)out_size; (void)ws_size;
    const float* a    = (const float*)d_in[0];   // (1, 8192, 128) fp32, L2-normalized
    const float* t    = (const float*)d_in[1];   // (1, 8192, 128) fp32
    const float* temp = (const float*)d_in[2];   // (1,1,1) log-temperature

    const int N = in_sizes[0] / D_DIM;           // 8192

    float* num   = (float*)d_ws;                 // N floats
    float* den_i = num + N;                      // N floats
    float* den_j = den_i + N;                    // N floats

    // workspace is poisoned by the harness -> zero the accumulators every call
    dcl_zero_kernel<<<(3 * N + 255) / 256, 256, 0, stream>>>(num, 3 * N);

    dim3 grid(N / BT, N / BT);                   // 64 x 64 tile blocks
    dcl_tile_kernel<<<grid, 256, 0, stream>>>(a, t, temp, num, den_i, den_j);

    dcl_finalize_kernel<<<1, 256, 0, stream>>>(num, den_i, den_j, (float*)d_out, N);
}